// RotatE_13013750907157
// MI455X (gfx1250) — compile-verified
//
#include <hip/hip_runtime.h>

// RotatE edge scoring for MI455X (gfx1250, wave32).
//
// score[t][e] = sum_c sqrt( (re_u*cr - im_u*sr - re_v)^2 + (im_u*cr + re_u*sr - im_v)^2 )
// with cr = cos(rel[c]/pi), sr = sin(rel[c]/pi), re/im = halves of x rows.
//
// Roofline: ~2.3 GFLOP total vs ~3 GB of gathered row reads; x (102 MB) fits
// in the 192 MB L2, so this is L2-bandwidth bound. One wave (32 lanes) per
// edge; each lane owns 4 components, so each half-row (128 floats) is exactly
// one coalesced 32-lane float4 load (global_load_b128). Next-iteration rows
// are prefetched with global_prefetch_b8. Output uses a non-temporal store so
// the write-once result stream does not evict the L2-resident x table.
//
// Trig is precomputed by a tiny kernel (3 x 2 x 128 values into d_ws): libm
// cosf/sinf carries the Payne-Hanek big-argument reduction and must not run
// per-block in the hot kernel.
//
// No WMMA: sqrt sits inside the per-edge reduction, so there is no matrix
// contraction to map onto v_wmma; forcing one would only add instructions to
// a memory-bound kernel. TDM/async-LDS staging likewise loses: the gathers
// are per-wave-private with VGPR-resident indices, and staging through LDS
// adds a round-trip with no bandwidth benefit.

constexpr int DIM_R = 128;   // components per half
constexpr int ROW   = 256;   // floats per x row
constexpr float INV_PI = 0.31830988618379067f;

__global__ __launch_bounds__(128) void rotate_trig_kernel(
    const float* __restrict__ rel0,
    const float* __restrict__ rel1,
    const float* __restrict__ rel2,
    float* __restrict__ tab)   // [3][2][128]: cos table then sin table per etype
{
    const float* rel = (blockIdx.x == 0) ? rel0 : (blockIdx.x == 1) ? rel1 : rel2;
    const int c = threadIdx.x;           // 0..127
    const float r = rel[c] * INV_PI;
    tab[blockIdx.x * 2 * DIM_R + c]         = cosf(r);
    tab[blockIdx.x * 2 * DIM_R + DIM_R + c] = sinf(r);
}

__global__ __launch_bounds__(256) void rotate_score_kernel(
    const float* __restrict__ x,
    const int*   __restrict__ u0, const int* __restrict__ v0,
    const int*   __restrict__ u1, const int* __restrict__ v1,
    const int*   __restrict__ u2, const int* __restrict__ v2,
    const float* __restrict__ tab,   // [3][2][128] cos/sin tables
    float*       __restrict__ out,   // [3][n_edges]
    int n_edges)
{
    // e-type select (uniform per block; a few SALU cselects at entry)
    const int et = blockIdx.y;
    const int* __restrict__ u = (et == 0) ? u0 : (et == 1) ? u1 : u2;
    const int* __restrict__ v = (et == 0) ? v0 : (et == 1) ? v1 : v2;
    const float* __restrict__ t = tab + et * 2 * DIM_R;
    float* __restrict__ o = out + (long long)et * n_edges;

    const int tid  = threadIdx.x;
    const int lane = tid & 31;   // wave32
    const int wav  = tid >> 5;   // 8 waves per block

    // Per-lane rotation coefficients for components [4*lane .. 4*lane+3].
    // Loaded once per wave from the (L2-resident) precomputed table.
    const float4 cr = reinterpret_cast<const float4*>(t)[lane];
    const float4 sr = reinterpret_cast<const float4*>(t + DIM_R)[lane];

    int e = blockIdx.x * 8 + wav;
    const int estride = gridDim.x * 8;

    for (; e < n_edges; e += estride) {
        const int uu = u[e];
        const int vv = v[e];

        // Prefetch both rows of the next edge this wave will process.
        // Each row is 1024 B = 8 x 128 B cachelines; lanes 0..7 cover the
        // u-row, lanes 8..15 the v-row.  -> global_prefetch_b8
        const int en = e + estride;
        if (en < n_edges) {
            const int un = u[en];
            const int vn = v[en];
            if (lane < 8) {
                __builtin_prefetch((const char*)(x + (size_t)un * ROW) + (size_t)lane * 128, 0, 3);
            } else if (lane < 16) {
                __builtin_prefetch((const char*)(x + (size_t)vn * ROW) + (size_t)(lane - 8) * 128, 0, 3);
            }
        }

        const float4* xu = reinterpret_cast<const float4*>(x + (size_t)uu * ROW);
        const float4* xv = reinterpret_cast<const float4*>(x + (size_t)vv * ROW);

        // Coalesced: 32 lanes x 16 B = one full 128-float row half per load.
        const float4 are = xu[lane];        // re_x[u], comps 4*lane..4*lane+3
        const float4 aim = xu[lane + 32];   // im_x[u]
        const float4 bre = xv[lane];        // re_x[v]
        const float4 bim = xv[lane + 32];   // im_x[v]

        float s = 0.0f;
        {
            float rs, is;
            rs = fmaf(are.x, cr.x, fmaf(-aim.x, sr.x, -bre.x));
            is = fmaf(aim.x, cr.x, fmaf( are.x, sr.x, -bim.x));
            s += sqrtf(fmaf(rs, rs, is * is));
            rs = fmaf(are.y, cr.y, fmaf(-aim.y, sr.y, -bre.y));
            is = fmaf(aim.y, cr.y, fmaf( are.y, sr.y, -bim.y));
            s += sqrtf(fmaf(rs, rs, is * is));
            rs = fmaf(are.z, cr.z, fmaf(-aim.z, sr.z, -bre.z));
            is = fmaf(aim.z, cr.z, fmaf( are.z, sr.z, -bim.z));
            s += sqrtf(fmaf(rs, rs, is * is));
            rs = fmaf(are.w, cr.w, fmaf(-aim.w, sr.w, -bre.w));
            is = fmaf(aim.w, cr.w, fmaf( are.w, sr.w, -bim.w));
            s += sqrtf(fmaf(rs, rs, is * is));
        }

        // wave32 butterfly reduction across the 32 lanes (5 steps)
        #pragma unroll
        for (int off = 16; off > 0; off >>= 1)
            s += __shfl_xor(s, off, 32);

        if (lane == 0)
            __builtin_nontemporal_store(s, &o[e]);   // write-once stream: TH=NT
    }
}

extern "C" void kernel_launch(void* const* d_in, const int* in_sizes, int n_in,
                              void* d_out, int out_size, void* d_ws, size_t ws_size,
                              hipStream_t stream) {
    // setup_inputs() order: x, u0, v0, u1, v1, u2, v2, rel0, rel1, rel2
    const float* x    = (const float*)d_in[0];
    const int*   u0   = (const int*)  d_in[1];
    const int*   v0   = (const int*)  d_in[2];
    const int*   u1   = (const int*)  d_in[3];
    const int*   v1   = (const int*)  d_in[4];
    const int*   u2   = (const int*)  d_in[5];
    const int*   v2   = (const int*)  d_in[6];
    const float* rel0 = (const float*)d_in[7];
    const float* rel1 = (const float*)d_in[8];
    const float* rel2 = (const float*)d_in[9];

    float* out = (float*)d_out;
    float* tab = (float*)d_ws;          // 3 * 256 floats = 3 KB of scratch
    const int E = in_sizes[1];          // 500000 edges per e-type

    // Precompute cos/sin(rel/pi) tables once per call (deterministic,
    // graph-capture safe: plain kernel launch on `stream`).
    rotate_trig_kernel<<<dim3(3), dim3(DIM_R), 0, stream>>>(rel0, rel1, rel2, tab);

    // One fused dispatch: grid.x strides edges (8 waves/block, 4 edges/wave),
    // grid.y = 3 e-types. 46875 blocks total -> single launch ramp/tail and
    // one big pool for the scheduler while x stays L2-resident.
    int blocks = (E + 8 * 4 - 1) / (8 * 4);
    dim3 grid(blocks, 3), block(256);

    rotate_score_kernel<<<grid, block, 0, stream>>>(x, u0, v0, u1, v1, u2, v2,
                                                    tab, out, E);
}